// EWConv_49632642073094
// MI455X (gfx1250) — compile-verified
//
#include <hip/hip_runtime.h>
#include <math.h>

// EWConv graph conv for MI455X (gfx1250, wave32).
// Phases: zero ws -> edge wsum/deg atomics -> WMMA GEMM (h) -> edge scatter -> WMMA GEMM (out, fused agg).

typedef __attribute__((ext_vector_type(2))) float v2f;
typedef __attribute__((ext_vector_type(8))) float v8f;

#define F 128  // IN_FEATS == OUT_FEATS == 128

__global__ void zero_kernel(float* __restrict__ p, int n) {
  int i = blockIdx.x * blockDim.x + threadIdx.x;
  if (i < n) p[i] = 0.0f;
}

__global__ void edge_prep_kernel(const float* __restrict__ efeat,
                                 const int* __restrict__ dst,
                                 float* __restrict__ wsum,
                                 float* __restrict__ deg, int nE) {
  int e = blockIdx.x * blockDim.x + threadIdx.x;
  if (e >= nE) return;
  int d = dst[e];
  atomicAdd(&wsum[d], efeat[e]);
  atomicAdd(&deg[d], 1.0f);
}

// One wave32 per edge; lane handles 4 consecutive feature channels (float4).
__global__ void __launch_bounds__(256)
edge_scatter_kernel(const float* __restrict__ efeat,
                    const int* __restrict__ src,
                    const int* __restrict__ dst,
                    const float* __restrict__ h,
                    const float* __restrict__ wsum,
                    float* __restrict__ agg, int nE) {
  int gid = blockIdx.x * blockDim.x + threadIdx.x;
  int e = gid >> 5;
  if (e >= nE) return;
  int lane = gid & 31;
  int s = src[e];
  int d = dst[e];
  float m = expf(-efeat[e] / wsum[d]);  // m = exp(-w_norm)
  const float4* hp = (const float4*)(h + (size_t)s * F);
  float4 hv = hp[lane];
  float* ap = agg + (size_t)d * F + lane * 4;
  atomicAdd(ap + 0, m * hv.x);
  atomicAdd(ap + 1, m * hv.y);
  atomicAdd(ap + 2, m * hv.z);
  atomicAdd(ap + 3, m * hv.w);
}

// out[m,n] = sum_k A[m,k] * W[n,k] + bias[n]  (+ agg[m,n] / max(deg[m],1) if fuse)
// One wave per 16-row tile; wave computes all 8 column tiles (full 128 cols).
// Uses V_WMMA_F32_16X16X4_F32, fp32 end to end (matches fp32 reference).
__global__ void __launch_bounds__(128)
gemm_wmma_kernel(const float* __restrict__ A,
                 const float* __restrict__ W,
                 const float* __restrict__ bias,
                 const float* __restrict__ agg,
                 const float* __restrict__ deg,
                 float* __restrict__ out,
                 int nTiles, int fuse) {
  // Stage W transposed in LDS: sWT[k*F + n] = W[n*F + k]  (64 KB)
  __shared__ float sWT[F * F];
  int tid = threadIdx.x;
  for (int i = tid; i < F * F; i += 128) {
    int n = i >> 7;
    int k = i & (F - 1);
    sWT[k * F + n] = W[i];
  }
  __syncthreads();

  int gw = blockIdx.x * 4 + (tid >> 5);
  if (gw >= nTiles) return;

  int lane = tid & 31;
  int half = lane >> 4;   // 0: lanes 0-15, 1: lanes 16-31
  int l16  = lane & 15;
  int m0   = gw * 16;
  int koff = half * 2;    // A/B fragment K sub-offset per ISA 16x4 f32 layout

  v8f acc[8] = {v8f{0.f}, v8f{0.f}, v8f{0.f}, v8f{0.f},
                v8f{0.f}, v8f{0.f}, v8f{0.f}, v8f{0.f}};

  const float* Arow = A + (size_t)(m0 + l16) * F;
  for (int kk = 0; kk < F; kk += 4) {
    v2f a;
    a.x = Arow[kk + koff];       // A[M=l16][K=kk+koff]
    a.y = Arow[kk + koff + 1];   // A[M=l16][K=kk+koff+1]
#pragma unroll
    for (int t = 0; t < 8; ++t) {
      int n = t * 16 + l16;
      v2f b;
      b.x = sWT[(kk + koff) * F + n];       // B[K][N=n] = W[n][K]
      b.y = sWT[(kk + koff + 1) * F + n];
      acc[t] = __builtin_amdgcn_wmma_f32_16x16x4_f32(
          false, a, false, b, (short)0, acc[t], false, false);
    }
  }

  // D layout: VGPR r -> M = r (lanes 0-15) / r+8 (lanes 16-31), N = l16
#pragma unroll
  for (int r = 0; r < 8; ++r) {
    int row = m0 + r + half * 8;
    float dmax = 1.0f;
    if (fuse) {
      float dv = deg[row];
      dmax = dv > 1.0f ? dv : 1.0f;
    }
#pragma unroll
    for (int t = 0; t < 8; ++t) {
      int col = t * 16 + l16;
      float v = acc[t][r] + bias[col];
      if (fuse) v += agg[(size_t)row * F + col] / dmax;
      out[(size_t)row * F + col] = v;
    }
  }
}

extern "C" void kernel_launch(void* const* d_in, const int* in_sizes, int n_in,
                              void* d_out, int out_size, void* d_ws, size_t ws_size,
                              hipStream_t stream) {
  (void)n_in; (void)out_size; (void)ws_size;
  const float* feat   = (const float*)d_in[0];
  const float* efeat  = (const float*)d_in[1];
  const int*   src    = (const int*)d_in[2];
  const int*   dst    = (const int*)d_in[3];
  const float* W_pool = (const float*)d_in[4];
  const float* b_pool = (const float*)d_in[5];
  const float* W_self = (const float*)d_in[6];
  const float* b_self = (const float*)d_in[7];
  float* out = (float*)d_out;

  int nN = in_sizes[0] / F;   // 50000
  int nE = in_sizes[2];       // 800000

  // Workspace layout (floats): h | agg | wsum | deg   (~51.6 MB total)
  float* h    = (float*)d_ws;
  float* agg  = h + (size_t)nN * F;
  float* wsum = agg + (size_t)nN * F;
  float* deg  = wsum + nN;

  // 1) zero agg + wsum + deg (contiguous)
  int nZero = nN * F + 2 * nN;
  zero_kernel<<<(nZero + 255) / 256, 256, 0, stream>>>(agg, nZero);

  // 2) per-edge wsum / deg
  edge_prep_kernel<<<(nE + 255) / 256, 256, 0, stream>>>(efeat, dst, wsum, deg, nE);

  // 3) h = feat @ W_pool^T + b_pool  (nN divisible by 16: 50000 = 3125*16)
  int nTiles = (nN + 15) / 16;
  int gemmBlocks = (nTiles + 3) / 4;
  gemm_wmma_kernel<<<gemmBlocks, 128, 0, stream>>>(feat, W_pool, b_pool,
                                                   agg, deg, h, nTiles, 0);

  // 4) scatter m * h[src] into agg[dst]
  long long sThreads = (long long)nE * 32;
  edge_scatter_kernel<<<(int)((sThreads + 255) / 256), 256, 0, stream>>>(
      efeat, src, dst, h, wsum, agg, nE);

  // 5) out = feat @ W_self^T + b_self + agg / max(deg,1)
  gemm_wmma_kernel<<<gemmBlocks, 128, 0, stream>>>(feat, W_self, b_self,
                                                   agg, deg, out, nTiles, 1);
}